// GraphSAGELayer_90890097918585
// MI455X (gfx1250) — compile-verified
//
#include <hip/hip_runtime.h>

typedef float v2f __attribute__((ext_vector_type(2)));
typedef float v8f __attribute__((ext_vector_type(8)));

#define N_NODES 100000
#define D 128
#define STRIPS (N_NODES / 16)   // 6250 row strips of 16

// ---------------------------------------------------------------------------
// Kernel 1: zero the workspace (agg [N*128] f32 followed by deg [N] f32)
// ---------------------------------------------------------------------------
__global__ void gsage_zero_kernel(float4* __restrict__ ws, int count4) {
    int i = blockIdx.x * blockDim.x + threadIdx.x;
    if (i < count4) ws[i] = make_float4(0.f, 0.f, 0.f, 0.f);
}

// ---------------------------------------------------------------------------
// Kernel 2: edge scatter. One wave32 per edge; each lane moves 4 consecutive
// floats (float4 coalesced load, 4 global f32 atomic adds into L2-resident agg).
// ---------------------------------------------------------------------------
__global__ void __launch_bounds__(256)
gsage_scatter_kernel(const float* __restrict__ x,
                     const int* __restrict__ adj,   // [2, E]: row0=src, row1=dst
                     float* __restrict__ agg,
                     float* __restrict__ deg,
                     int E) {
    int e    = blockIdx.x * 8 + (threadIdx.x >> 5);
    int lane = threadIdx.x & 31;
    if (e >= E) return;

    int src = adj[e];
    int dst = adj[E + e];

    const float4* xr = (const float4*)(x + (size_t)dst * D);
    float4 v = xr[lane];                       // 512B coalesced per wave

    float* a = agg + (size_t)src * D + lane * 4;
    atomicAdd(a + 0, v.x);
    atomicAdd(a + 1, v.y);
    atomicAdd(a + 2, v.z);
    atomicAdd(a + 3, v.w);
    if (lane == 0) atomicAdd(deg + src, 1.0f);
}

// ---------------------------------------------------------------------------
// Kernel 3: out = (agg * 1/max(deg,1)) @ W using V_WMMA_F32_16X16X4_F32.
// Block = 256 threads = 8 waves; wave w owns output columns [16w, 16w+16).
// B fragment (its 16 columns, all K=128 -> 64 VGPRs) loaded once, reused
// across grid-strided row strips. A strip staged (degree-scaled) in LDS.
// ---------------------------------------------------------------------------
__global__ void __launch_bounds__(256)
gsage_wmma_gemm_kernel(const float* __restrict__ agg,
                       const float* __restrict__ deg,
                       const float* __restrict__ W,   // [D_IN=128, D_OUT=128] row-major
                       float* __restrict__ out) {
    __shared__ float As[16][D + 2];   // +2 pad: lane->bank stride 130 (=2 mod 64)
    __shared__ float rd[16];

    const int tid  = threadIdx.x;
    const int wave = tid >> 5;
    const int lane = tid & 31;
    const int n0   = wave * 16;
    const int koff = (lane >> 4) * 2;   // lanes 16-31 hold K pair +2
    const int nn   = lane & 15;

    // --- Preload this wave's B fragments for all 32 K-steps (f32 B 4x16:
    //     lane = column n, VGPR0/1 = K koff, koff+1)
    v2f breg[32];
#pragma unroll
    for (int s = 0; s < 32; ++s) {
        int k = 4 * s + koff;
        breg[s].x = W[(size_t)k       * D + n0 + nn];
        breg[s].y = W[(size_t)(k + 1) * D + n0 + nn];
    }

    const int mA = lane & 15;   // A row held by this lane

    for (int strip = blockIdx.x; strip < STRIPS; strip += gridDim.x) {
        const int m0 = strip * 16;

        __syncthreads();   // protect As/rd from previous iteration's readers
        if (tid < 16) {
            float d = deg[m0 + tid];
            rd[tid] = 1.0f / fmaxf(d, 1.0f);
        }
        __syncthreads();

        // Stage degree-normalized A strip: 16x128 floats, 8 per thread.
#pragma unroll
        for (int j = 0; j < 8; ++j) {
            int i = tid + 256 * j;
            int r = i >> 7;        // row 0..15
            int c = i & (D - 1);   // col 0..127
            As[r][c] = agg[(size_t)(m0 + r) * D + c] * rd[r];
        }
        __syncthreads();

        v8f acc = {0.f, 0.f, 0.f, 0.f, 0.f, 0.f, 0.f, 0.f};
#pragma unroll
        for (int s = 0; s < 32; ++s) {
            int k = 4 * s + koff;
            v2f a;
            a.x = As[mA][k];
            a.y = As[mA][k + 1];
            // D = A(16x4) * B(4x16) + C, full fp32
            acc = __builtin_amdgcn_wmma_f32_16x16x4_f32(
                false, a, false, breg[s], (short)0, acc, false, false);
        }

        // C/D layout: VGPR r -> M = r (lanes 0-15) / r+8 (lanes 16-31), N = lane&15
        const int mbase = m0 + ((lane >> 4) << 3);
#pragma unroll
        for (int r = 0; r < 8; ++r) {
            out[(size_t)(mbase + r) * D + n0 + nn] = acc[r];
        }
    }
}

// ---------------------------------------------------------------------------
extern "C" void kernel_launch(void* const* d_in, const int* in_sizes, int n_in,
                              void* d_out, int out_size, void* d_ws, size_t ws_size,
                              hipStream_t stream) {
    const float* x   = (const float*)d_in[0];   // [N, 128] f32
    const int*   adj = (const int*)d_in[1];     // [2, E] int
    const float* W   = (const float*)d_in[2];   // [128, 128] f32
    float*       out = (float*)d_out;           // [N, 128] f32

    const int E = in_sizes[1] / 2;

    float* agg = (float*)d_ws;                       // N*128 f32
    float* deg = agg + (size_t)N_NODES * D;          // N f32

    // 1) zero accumulators
    const int count4 = (N_NODES * (D + 1)) / 4;      // agg+deg are contiguous
    gsage_zero_kernel<<<(count4 + 255) / 256, 256, 0, stream>>>((float4*)d_ws, count4);

    // 2) edge scatter (one wave per edge)
    gsage_scatter_kernel<<<(E + 7) / 8, 256, 0, stream>>>(x, adj, agg, deg, E);

    // 3) normalize + GEMM via f32 WMMA (each block handles ~10 row strips)
    gsage_wmma_gemm_kernel<<<640, 256, 0, stream>>>(agg, deg, W, out);
}